// Model2_Transformer_63264868270719
// MI455X (gfx1250) — compile-verified
//
#include <hip/hip_runtime.h>
#include <math.h>

#define NF 16384
#define EF 131072
#define NG 64
#define EDIM 7

typedef float v2f __attribute__((ext_vector_type(2)));
typedef float v8f __attribute__((ext_vector_type(8)));
typedef int   v4i __attribute__((ext_vector_type(4)));

#if __has_builtin(__builtin_amdgcn_global_load_async_to_lds_b128) && \
    __has_builtin(__builtin_amdgcn_s_wait_asynccnt)
#define USE_ASYNC_LDS 1
typedef __attribute__((address_space(1))) v4i g_v4i;   // global src
typedef __attribute__((address_space(3))) v4i l_v4i;   // LDS dst
#else
#define USE_ASYNC_LDS 0
#endif

__device__ __forceinline__ float gelu_exact(float x) {
  return 0.5f * x * (1.0f + erff(x * 0.70710678118654752440f));
}

// ---------------- WMMA f32 GEMM:  C[M,N] = A[M,K] @ W[K,N] + bias ----------------
// block = 256 threads = 8 waves; each wave owns one 16x16 C tile; block tile 32x64.
// Full interior tiles are staged via async global->LDS b128 (ASYNCcnt path);
// ragged edge tiles use guarded, zero-padded scalar staging.
#define TM 32
#define TN 64
#define TK 32
#define APAD 4  // keep 16B alignment for b128 LDS writes; 36-float stride -> 16 banks

__global__ __launch_bounds__(256) void gemm_bias_kernel(
    const float* __restrict__ A, const float* __restrict__ Wt,
    const float* __restrict__ bias, float* __restrict__ C,
    int M, int K, int N)
{
  __shared__ float As[TM][TK + APAD];
  __shared__ float Bs[TK][TN + APAD];
  const int tid  = threadIdx.x;
  const int lane = tid & 31;
  const int wave = tid >> 5;
  const int wr   = wave >> 2;            // 0..1 (row of 16)
  const int wc   = wave & 3;             // 0..3 (col of 16)
  const int l16  = lane & 15;
  const int koff = (lane >> 4) << 1;     // lanes 16..31 handle K+2,K+3
  const int bm = blockIdx.y * TM;
  const int bn = blockIdx.x * TN;

  v8f acc = {0.f, 0.f, 0.f, 0.f, 0.f, 0.f, 0.f, 0.f};

  for (int k0 = 0; k0 < K; k0 += TK) {
    const bool full = (k0 + TK <= K) && (bn + TN <= N) && (bm + TM <= M);
#if USE_ASYNC_LDS
    if (full) {
      {   // A tile: 32 rows x 8 x 16B segments = 256 b128 ops, one per thread
        int r = tid >> 3, c = (tid & 7) << 2;
        __builtin_amdgcn_global_load_async_to_lds_b128(
            (g_v4i*)(A + (size_t)(bm + r) * K + (k0 + c)),
            (l_v4i*)&As[r][c], 0, 0);
      }
#pragma unroll
      for (int i = 0; i < 2; ++i) {  // B tile: 32 rows x 16 segments = 512 b128 ops
        int t = tid + i * 256;
        int r = t >> 4, c = (t & 15) << 2;
        __builtin_amdgcn_global_load_async_to_lds_b128(
            (g_v4i*)(Wt + (size_t)(k0 + r) * N + (bn + c)),
            (l_v4i*)&Bs[r][c], 0, 0);
      }
      __builtin_amdgcn_s_wait_asynccnt(0);
    } else
#endif
    {
      for (int i = tid; i < TM * TK; i += 256) {
        int r = i >> 5, c = i & (TK - 1);
        int gm = bm + r, gk = k0 + c;
        As[r][c] = (gm < M && gk < K) ? A[(size_t)gm * K + gk] : 0.0f;
      }
      for (int i = tid; i < TK * TN; i += 256) {
        int r = i >> 6, c = i & (TN - 1);
        int gk = k0 + r, gn = bn + c;
        Bs[r][c] = (gk < K && gn < N) ? Wt[(size_t)gk * N + gn] : 0.0f;
      }
    }
    __syncthreads();
#pragma unroll
    for (int kk = 0; kk < TK; kk += 4) {
      v2f a, b;
      a.x = As[wr * 16 + l16][kk + koff];
      a.y = As[wr * 16 + l16][kk + koff + 1];
      b.x = Bs[kk + koff][wc * 16 + l16];
      b.y = Bs[kk + koff + 1][wc * 16 + l16];
      acc = __builtin_amdgcn_wmma_f32_16x16x4_f32(false, a, false, b,
                                                  (short)0, acc, false, false);
    }
    __syncthreads();
  }

  // C/D layout: VGPR r -> row (r) lanes 0-15, row (r+8) lanes 16-31; col = lane%16
  const int col = bn + wc * 16 + l16;
  const int rb  = bm + wr * 16 + ((lane >> 4) << 3);
  if (col < N) {
    float bv = bias ? bias[col] : 0.0f;
#pragma unroll
    for (int r = 0; r < 8; ++r) {
      int row = rb + r;
      if (row < M) C[(size_t)row * N + col] = acc[r] + bv;
    }
  }
}

// ---------------- utility kernels ----------------
__global__ void fill_kernel(float* __restrict__ p, float v, long n) {
  long i = (long)blockIdx.x * blockDim.x + threadIdx.x;
  if (i < n) p[i] = v;
}

__device__ __forceinline__ void atomicMaxF(float* addr, float val) {
  unsigned int* ua = (unsigned int*)addr;
  unsigned int old = *ua;
  while (__uint_as_float(old) < val) {
    unsigned int assumed = old;
    old = atomicCAS(ua, assumed, __float_as_uint(val));
    if (old == assumed) break;
  }
}

// ---------------- edge kernels (wave per (edge, head)) ----------------
__global__ __launch_bounds__(256) void edge_alpha_kernel(
    const float* __restrict__ Q, const float* __restrict__ Kf,
    const float* __restrict__ EA, const float* __restrict__ WE,
    const int* __restrict__ src, const int* __restrict__ dst,
    float* __restrict__ alpha, int E, int H, int C, float scale)
{
  int gw   = (int)((blockIdx.x * 256u + threadIdx.x) >> 5);
  int lane = threadIdx.x & 31;
  if (gw >= E * H) return;
  int e = gw / H, h = gw - e * H;
  int s = src[e], d = dst[e];
  int HC = H * C;
  float ea[EDIM];
#pragma unroll
  for (int j = 0; j < EDIM; ++j) ea[j] = EA[(size_t)e * EDIM + j];
  float acc = 0.0f;
  for (int c = lane; c < C; c += 32) {
    int f = h * C + c;
    float ep = 0.0f;
#pragma unroll
    for (int j = 0; j < EDIM; ++j) ep += ea[j] * WE[(size_t)j * HC + f];
    acc += Q[(size_t)d * HC + f] * (Kf[(size_t)s * HC + f] + ep);
  }
#pragma unroll
  for (int off = 16; off > 0; off >>= 1) acc += __shfl_xor(acc, off, 32);
  if (lane == 0) alpha[(size_t)e * H + h] = acc * scale;
}

__global__ void seg_max_kernel(const float* __restrict__ alpha,
                               const int* __restrict__ dst,
                               float* __restrict__ amax, int E, int H)
{
  int i = blockIdx.x * blockDim.x + threadIdx.x;
  if (i >= E * H) return;
  int e = i / H, h = i - e * H;
  atomicMaxF(&amax[(size_t)dst[e] * H + h], alpha[i]);
}

__global__ void fix_amax_kernel(float* __restrict__ amax, long n) {
  long i = (long)blockIdx.x * blockDim.x + threadIdx.x;
  if (i < n) {
    float v = amax[i];
    if (!isfinite(v)) amax[i] = 0.0f;
  }
}

__global__ void exp_den_kernel(float* __restrict__ alpha_ex,
                               const float* __restrict__ amax,
                               const int* __restrict__ dst,
                               float* __restrict__ denom, int E, int H)
{
  int i = blockIdx.x * blockDim.x + threadIdx.x;
  if (i >= E * H) return;
  int e = i / H, h = i - e * H;
  size_t di = (size_t)dst[e] * H + h;
  float ex = expf(alpha_ex[i] - amax[di]);
  alpha_ex[i] = ex;
  atomicAdd(&denom[di], ex);
}

__global__ __launch_bounds__(256) void msg_scatter_kernel(
    const float* __restrict__ V, const float* __restrict__ EA,
    const float* __restrict__ WE, const float* __restrict__ ex,
    const float* __restrict__ denom,
    const int* __restrict__ src, const int* __restrict__ dst,
    float* __restrict__ agg, int E, int H, int C)
{
  int gw   = (int)((blockIdx.x * 256u + threadIdx.x) >> 5);
  int lane = threadIdx.x & 31;
  if (gw >= E * H) return;
  int e = gw / H, h = gw - e * H;
  int s = src[e], d = dst[e];
  int HC = H * C;
  float a = ex[(size_t)e * H + h] / (denom[(size_t)d * H + h] + 1e-16f);
  float ea[EDIM];
#pragma unroll
  for (int j = 0; j < EDIM; ++j) ea[j] = EA[(size_t)e * EDIM + j];
  for (int c = lane; c < C; c += 32) {
    int f = h * C + c;
    float ep = 0.0f;
#pragma unroll
    for (int j = 0; j < EDIM; ++j) ep += ea[j] * WE[(size_t)j * HC + f];
    atomicAdd(&agg[(size_t)d * HC + f], (V[(size_t)s * HC + f] + ep) * a);
  }
}

// ---------------- combine / pool / head ----------------
__global__ void combine_concat_kernel(const float* __restrict__ agg,
                                      const float* __restrict__ S,
                                      float* __restrict__ out, long n)
{
  long i = (long)blockIdx.x * blockDim.x + threadIdx.x;
  if (i < n) out[i] = gelu_exact(agg[i] + S[i]);
}

__global__ void combine_mean_kernel(const float* __restrict__ agg,
                                    const float* __restrict__ S,
                                    float* __restrict__ out, int N, int H, int C)
{
  int i = blockIdx.x * blockDim.x + threadIdx.x;
  if (i >= N * C) return;
  int n = i / C, c = i - n * C;
  float m = 0.0f;
  for (int h = 0; h < H; ++h) m += agg[(size_t)n * H * C + h * C + c];
  m /= (float)H;
  out[i] = gelu_exact(m + S[i]);
}

__global__ void pool_kernel(const float* __restrict__ h, float* __restrict__ pooled,
                            int npg, int F)
{
  int g = blockIdx.x, f = threadIdx.x;
  if (f >= F) return;
  float s = 0.0f, mx = -INFINITY;
  long base = (long)g * npg;
  for (int n = 0; n < npg; ++n) {
    float v = h[(base + n) * F + f];
    s += v;
    mx = fmaxf(mx, v);
  }
  pooled[(size_t)g * 3 * F + f]         = s / (float)npg;
  pooled[(size_t)g * 3 * F + F + f]     = mx;
  pooled[(size_t)g * 3 * F + 2 * F + f] = s;
}

__global__ __launch_bounds__(256) void gelu_ln_kernel(
    float* __restrict__ z, const float* __restrict__ g,
    const float* __restrict__ b, int D)
{
  __shared__ float red[256];
  float* zr = z + (size_t)blockIdx.x * D;
  float s = 0.0f;
  for (int i = threadIdx.x; i < D; i += 256) {
    float v = gelu_exact(zr[i]);
    zr[i] = v;
    s += v;
  }
  red[threadIdx.x] = s;
  __syncthreads();
  for (int o = 128; o > 0; o >>= 1) {
    if (threadIdx.x < o) red[threadIdx.x] += red[threadIdx.x + o];
    __syncthreads();
  }
  float mean = red[0] / (float)D;
  __syncthreads();
  float vs = 0.0f;
  for (int i = threadIdx.x; i < D; i += 256) {
    float d2 = zr[i] - mean;
    vs += d2 * d2;
  }
  red[threadIdx.x] = vs;
  __syncthreads();
  for (int o = 128; o > 0; o >>= 1) {
    if (threadIdx.x < o) red[threadIdx.x] += red[threadIdx.x + o];
    __syncthreads();
  }
  float inv = rsqrtf(red[0] / (float)D + 1e-5f);
  __syncthreads();
  for (int i = threadIdx.x; i < D; i += 256)
    zr[i] = (zr[i] - mean) * inv * g[i] + b[i];
}

// ---------------- host orchestration ----------------
extern "C" void kernel_launch(void* const* d_in, const int* in_sizes, int n_in,
                              void* d_out, int out_size, void* d_ws, size_t ws_size,
                              hipStream_t stream)
{
  (void)n_in; (void)out_size; (void)ws_size;
  auto F = [&](int i) { return (const float*)d_in[i]; };

  const float* cw[4][9];                 // wq,bq,wk,bk,wv,bv,we,ws,bs
  const float *hw[5], *hb[5], *lg[4], *lbp[4];
  const float *xp, *eap;
  const int *eip;

  bool insertion = (in_sizes[0] == NF * 14);
  if (insertion) {
    // top-level dict insertion order: x, edge_attr, params(conv0..3, head, lns), edge_index, batch
    xp = F(0); eap = F(1);
    for (int i = 0; i < 4; ++i)
      for (int j = 0; j < 9; ++j) cw[i][j] = F(2 + 9 * i + j);
    for (int j = 0; j < 5; ++j) { hw[j] = F(38 + 2 * j); hb[j] = F(39 + 2 * j); }
    for (int j = 0; j < 4; ++j) { lg[j] = F(48 + 2 * j); lbp[j] = F(49 + 2 * j); }
    eip = (const int*)d_in[56];
  } else {
    // jax tree-flatten order (dict keys sorted): batch, edge_attr, edge_index, params, x
    eap = F(1); eip = (const int*)d_in[2];
    for (int i = 0; i < 4; ++i) {
      int b0 = 3 + 9 * i;                // sorted keys: bk,bq,bs,bv,we,wk,wq,ws,wv
      cw[i][0] = F(b0 + 6);  // wq
      cw[i][1] = F(b0 + 1);  // bq
      cw[i][2] = F(b0 + 5);  // wk
      cw[i][3] = F(b0 + 0);  // bk
      cw[i][4] = F(b0 + 8);  // wv
      cw[i][5] = F(b0 + 3);  // bv
      cw[i][6] = F(b0 + 4);  // we
      cw[i][7] = F(b0 + 7);  // ws
      cw[i][8] = F(b0 + 2);  // bs
    }
    for (int j = 0; j < 5; ++j) { hb[j] = F(39 + 2 * j); hw[j] = F(40 + 2 * j); }
    for (int j = 0; j < 4; ++j) { lbp[j] = F(49 + 2 * j); lg[j] = F(50 + 2 * j); }
    xp = F(57);
  }
  const int* srcp = eip;
  const int* dstp = eip + EF;

  // workspace layout (floats)
  float* ws = (float*)d_ws;
  size_t o = 0;
  auto take = [&](size_t n) { float* p = ws + o; o += n; return p; };
  float* Qb = take((size_t)NF * 768);
  float* Kb = take((size_t)NF * 768);
  float* Vb = take((size_t)NF * 768);
  float* Sb = take((size_t)NF * 768);
  float* AG = take((size_t)NF * 768);
  float* HA = take((size_t)NF * 768);
  float* HB = take((size_t)NF * 768);
  float* EXb = take((size_t)EF * 6);
  float* AM  = take((size_t)NF * 6);
  float* DN  = take((size_t)NF * 6);
  float* PL  = take((size_t)NG * 192);
  float* ZA  = take((size_t)NG * 320);
  float* ZB  = take((size_t)NG * 320);

  auto GEMM = [&](const float* A, const float* Wt, const float* b, float* Cn,
                  int M, int K, int N) {
    dim3 g((unsigned)((N + TN - 1) / TN), (unsigned)((M + TM - 1) / TM));
    gemm_bias_kernel<<<g, 256, 0, stream>>>(A, Wt, b, Cn, M, K, N);
  };
  auto FILL = [&](float* p, float v, long n) {
    fill_kernel<<<(unsigned)((n + 255) / 256), 256, 0, stream>>>(p, v, n);
  };

  struct CConf { int fin, H, C, concat; };
  const CConf convs[4] = {{14, 6, 128, 1}, {768, 6, 128, 1}, {768, 4, 64, 1}, {256, 2, 64, 0}};

  const float* hcur = xp;
  float* houts[2] = {HA, HB};
  for (int i = 0; i < 4; ++i) {
    const int fin = convs[i].fin, H = convs[i].H, C = convs[i].C, cc = convs[i].concat;
    const int HC = H * C, SK = cc ? HC : C;
    const float* const* p = cw[i];
    GEMM(hcur, p[0], p[1], Qb, NF, fin, HC);
    GEMM(hcur, p[2], p[3], Kb, NF, fin, HC);
    GEMM(hcur, p[4], p[5], Vb, NF, fin, HC);
    GEMM(hcur, p[7], p[8], Sb, NF, fin, SK);
    FILL(AM, -INFINITY, (long)NF * H);
    FILL(DN, 0.0f, (long)NF * H);
    FILL(AG, 0.0f, (long)NF * HC);
    const int EH = EF * H;
    const float scale = 1.0f / sqrtf((float)C);
    edge_alpha_kernel<<<(EH + 7) / 8, 256, 0, stream>>>(Qb, Kb, eap, p[6], srcp, dstp,
                                                        EXb, EF, H, C, scale);
    seg_max_kernel<<<(EH + 255) / 256, 256, 0, stream>>>(EXb, dstp, AM, EF, H);
    fix_amax_kernel<<<(unsigned)(((long)NF * H + 255) / 256), 256, 0, stream>>>(AM, (long)NF * H);
    exp_den_kernel<<<(EH + 255) / 256, 256, 0, stream>>>(EXb, AM, dstp, DN, EF, H);
    msg_scatter_kernel<<<(EH + 7) / 8, 256, 0, stream>>>(Vb, eap, p[6], EXb, DN,
                                                         srcp, dstp, AG, EF, H, C);
    float* hn = houts[i & 1];
    if (cc)
      combine_concat_kernel<<<(unsigned)(((long)NF * HC + 255) / 256), 256, 0, stream>>>(
          AG, Sb, hn, (long)NF * HC);
    else
      combine_mean_kernel<<<(NF * C + 255) / 256, 256, 0, stream>>>(AG, Sb, hn, NF, H, C);
    hcur = hn;
  }

  // pooling: [64, 192] = concat(mean, max, sum) over 256 nodes per graph, F=64
  pool_kernel<<<NG, 64, 0, stream>>>(hcur, PL, NF / NG, 64);

  // MLP head
  const int hd[6] = {192, 320, 192, 96, 48, 9};
  float* zbuf[2] = {ZA, ZB};
  const float* zin = PL;
  for (int j = 0; j < 4; ++j) {
    float* zo = zbuf[j & 1];
    GEMM(zin, hw[j], hb[j], zo, NG, hd[j], hd[j + 1]);
    gelu_ln_kernel<<<NG, 256, 0, stream>>>(zo, lg[j], lbp[j], hd[j + 1]);
    zin = zo;
  }
  GEMM(zin, hw[4], hb[4], (float*)d_out, NG, hd[4], hd[5]);
}